// MHC_27058293965527
// MI455X (gfx1250) — compile-verified
//
#include <hip/hip_runtime.h>
#include <math.h>

// ---------------------------------------------------------------------------
// MHC fused head kernel for MI455X (gfx1250, wave32, WMMA).
//   x (B,T,4096) = transpose of stream(B,4,T,1024); one pass over stream:
//   fp32 WMMA GEMM  y = x @ (W .* norm_w)^T   (24 outputs, padded to 32)
//   + fused sum-of-squares for RMSNorm, epilogue: sigmoid / 2*sigmoid /
//   exp + 6 Sinkhorn-Knopp iterations on 4x4, float4 stores.
// ---------------------------------------------------------------------------

typedef float v2f __attribute__((ext_vector_type(2)));
typedef float v8f __attribute__((ext_vector_type(8)));

#define THREADS     256
#define WAVES       8
#define TOK_PER_WG  128          // 8 waves * 16 tokens
#define KCHUNK      256          // K staged per LDS chunk (== THREADS)
#define ND          4096         // n*d
#define T_DIM       4096
#define D_DIM       1024
#define N_HEADS     4
#define NOUT_PAD    32           // 4 pre + 4 post + 16 res + 8 zero pad
#define BROW        96           // floats per k-pair row in LDS (48 float2); bank-friendly
#define YS          34           // epilogue y stride (conflict-free: gcd(34,64)=2, 17 odd)
#define EPS_RMS     1.1920928955078125e-07f

__global__ __launch_bounds__(THREADS)
void mhc_fused(const float* __restrict__ stream_in,
               const float* __restrict__ norm_w,
               const float* __restrict__ w_pre,
               const float* __restrict__ w_post,
               const float* __restrict__ w_res,
               const float* __restrict__ b_pre,
               const float* __restrict__ b_post,
               const float* __restrict__ b_res,
               const float* __restrict__ a_pre_p,
               const float* __restrict__ a_post_p,
               const float* __restrict__ a_res_p,
               float* __restrict__ out,
               int total_tokens)
{
    // 128 k-pairs * 96 floats = 49152 B; reused by the epilogue stage.
    __shared__ float smem[(KCHUNK / 2) * BROW];

    const int tid  = threadIdx.x;
    const int wave = tid >> 5;
    const int lane = tid & 31;
    const int col  = lane & 15;   // A-row token index / D-column index
    const int half = lane >> 4;   // 0: K 0..1, 1: K 2..3 of each step

    const int wgTok   = blockIdx.x * TOK_PER_WG;
    const int tileTok = wgTok + wave * 16;
    const int bIdx    = tileTok >> 12;          // / T_DIM
    const int t0      = tileTok & (T_DIM - 1);

    v8f acc0 = {0.f, 0.f, 0.f, 0.f, 0.f, 0.f, 0.f, 0.f};  // outputs 0..15
    v8f acc1 = {0.f, 0.f, 0.f, 0.f, 0.f, 0.f, 0.f, 0.f};  // outputs 16..31
    float ss = 0.f;                                        // partial sum(x^2)

    for (int c = 0; c < ND / KCHUNK; ++c) {
        const int k0 = c * KCHUNK;

        // ---- cooperative load: B_lds[k][n] = W[n][k0+k] * norm_w[k0+k] ----
        {
            const int   kl = tid;            // KCHUNK == THREADS
            const int   kg = k0 + kl;
            const float nw = norm_w[kg];
            float* dst = &smem[(kl >> 1) * BROW + (kl & 1)];
#pragma unroll
            for (int n = 0; n < NOUT_PAD; ++n) {
                float v = 0.f;
                if (n < 4)       v = w_pre [(size_t)n       * ND + kg];
                else if (n < 8)  v = w_post[(size_t)(n - 4) * ND + kg];
                else if (n < 24) v = w_res [(size_t)(n - 8) * ND + kg];
                dst[n * 2] = v * nw;
            }
        }
        __syncthreads();

        // ---- A-tile base for this chunk (x row is 4 strided d-segments) ----
        const int nseg = k0 >> 10;           // which head segment of x-row
        const int koff = k0 & (D_DIM - 1);
        const float* Ap = stream_in
            + (size_t)(bIdx * N_HEADS + nseg) * (size_t)(T_DIM * D_DIM)
            + (size_t)(t0 + col) * D_DIM + koff + half * 2;

#pragma unroll 4
        for (int kk = 0; kk < KCHUNK; kk += 4) {
            v2f a = *(const v2f*)(Ap + kk);                   // global_load_b64
            ss = fmaf(a.x, a.x, ss);
            ss = fmaf(a.y, a.y, ss);
            const float* bp = &smem[((kk >> 1) + half) * BROW + col * 2];
            v2f b0 = *(const v2f*)(bp);                       // ds_load_b64
            v2f b1 = *(const v2f*)(bp + 32);                  // cols 16..31
            acc0 = __builtin_amdgcn_wmma_f32_16x16x4_f32(
                       false, a, false, b0, (short)0, acc0, false, false);
            acc1 = __builtin_amdgcn_wmma_f32_16x16x4_f32(
                       false, a, false, b1, (short)0, acc1, false, false);
        }
        __syncthreads();
    }

    // ---- stage y and sum-of-squares through LDS (weights are dead now) ----
    float* ylds  = smem;                         // [TOK_PER_WG][YS]
    float* sslds = smem + TOK_PER_WG * YS;       // [TOK_PER_WG]

    const int tb = wave * 16;
#pragma unroll
    for (int j = 0; j < 8; ++j) {
        const int row = tb + j + half * 8;       // token within workgroup
        ylds[row * YS + col]      = acc0[j];
        ylds[row * YS + col + 16] = acc1[j];
    }
    const float ssF = ss + __shfl_xor(ss, 16, 32);
    if (half == 0) sslds[tb + col] = ssF;
    __syncthreads();

    // ---- per-token epilogue: RMS scale, heads, Sinkhorn-Knopp ----
    if (tid < TOK_PER_WG) {
        const int   g      = wgTok + tid;
        const float a_pre  = *a_pre_p;
        const float a_post = *a_post_p;
        const float a_res  = *a_res_p;
        const float scale  = rsqrtf(sslds[tid] * (1.0f / (float)ND) + EPS_RMS);
        const float* yr    = &ylds[tid * YS];

        float pre[4], post[4];
#pragma unroll
        for (int i = 0; i < 4; ++i) {
            const float zp = fmaf(a_pre * scale,  yr[i],     b_pre[i]);
            pre[i]  = 1.0f / (1.0f + __expf(-zp));
            const float zq = fmaf(a_post * scale, yr[4 + i], b_post[i]);
            post[i] = 2.0f / (1.0f + __expf(-zq));
        }

        float Mv[16];
#pragma unroll
        for (int m = 0; m < 16; ++m)
            Mv[m] = __expf(fmaf(a_res * scale, yr[8 + m], b_res[m]));

#pragma unroll
        for (int it = 0; it < 6; ++it) {
#pragma unroll
            for (int i = 0; i < 4; ++i) {       // row normalize (axis=-1)
                const float s = Mv[i*4+0] + Mv[i*4+1] + Mv[i*4+2] + Mv[i*4+3];
                const float r = 1.0f / s;
                Mv[i*4+0] *= r; Mv[i*4+1] *= r; Mv[i*4+2] *= r; Mv[i*4+3] *= r;
            }
#pragma unroll
            for (int j2 = 0; j2 < 4; ++j2) {    // col normalize (axis=-2)
                const float s = Mv[j2] + Mv[4+j2] + Mv[8+j2] + Mv[12+j2];
                const float r = 1.0f / s;
                Mv[j2] *= r; Mv[4+j2] *= r; Mv[8+j2] *= r; Mv[12+j2] *= r;
            }
        }

        // ---- outputs: H_res (B,T,4,4), then H_pre (B,T,4), then H_post ----
        float4* res4 = (float4*)(out + (size_t)g * 16);
        res4[0] = make_float4(Mv[0],  Mv[1],  Mv[2],  Mv[3]);
        res4[1] = make_float4(Mv[4],  Mv[5],  Mv[6],  Mv[7]);
        res4[2] = make_float4(Mv[8],  Mv[9],  Mv[10], Mv[11]);
        res4[3] = make_float4(Mv[12], Mv[13], Mv[14], Mv[15]);

        const size_t NTOK = (size_t)total_tokens;
        *(float4*)(out + NTOK * 16 + (size_t)g * 4) =
            make_float4(pre[0], pre[1], pre[2], pre[3]);
        *(float4*)(out + NTOK * 20 + (size_t)g * 4) =
            make_float4(post[0], post[1], post[2], post[3]);
    }
}

extern "C" void kernel_launch(void* const* d_in, const int* in_sizes, int n_in,
                              void* d_out, int out_size, void* d_ws, size_t ws_size,
                              hipStream_t stream)
{
    const float* x_stream = (const float*)d_in[0];
    const float* norm_w   = (const float*)d_in[1];
    const float* w_pre    = (const float*)d_in[2];
    const float* w_post   = (const float*)d_in[3];
    const float* w_res    = (const float*)d_in[4];
    const float* b_pre    = (const float*)d_in[5];
    const float* b_post   = (const float*)d_in[6];
    const float* b_res    = (const float*)d_in[7];
    const float* a_pre    = (const float*)d_in[8];
    const float* a_post   = (const float*)d_in[9];
    const float* a_res    = (const float*)d_in[10];

    const int nd    = in_sizes[1];            // n*d = 4096
    const int total = in_sizes[0] / nd;       // B*T  = 16384 tokens

    dim3 grid(total / TOK_PER_WG);
    dim3 block(THREADS);
    mhc_fused<<<grid, block, 0, stream>>>(x_stream, norm_w, w_pre, w_post, w_res,
                                          b_pre, b_post, b_res,
                                          a_pre, a_post, a_res,
                                          (float*)d_out, total);
}